// SparseGeometricInteraction_66803921322217
// MI455X (gfx1250) — compile-verified
//
#include <hip/hip_runtime.h>
#include <hip/hip_bf16.h>

typedef __attribute__((ext_vector_type(2))) float v2f;
typedef __attribute__((ext_vector_type(8))) float v8f;

// ---------------------------------------------------------------------------
// f32 WMMA GEMM with LDS-staged B: C[M x Nc] = A[M x K] * B[K x Nc] (+ReLU)
//
// Block = 256 threads (8 waves) computing a 128(M) x 128(N) tile; wave w owns
// rows [tile + 16w, tile + 16w + 16) and all 128 columns (8 accumulators).
// K is chunked by 32: the 32x128 B chunk is staged cooperatively into LDS
// *transposed* (Bs[col][k], row stride 34 so ds_load_b64 of (k,k+1) is
// 8B-aligned and bank-conflict-free), then each wave runs 8 K-steps of
// V_WMMA_F32_16X16X4_F32: one global v2f A load amortized over 8 WMMAs,
// B fragments via ds_load_b64 (pair-merged to ds_load_2addr_b64).
//
// B addressing:
//   bBlkShift >= 0 : column c -> block (c>>shift), col (c & mask);
//                    element = B[blk*bBlkStride + k*ldb + col]   (enc_q/enc_v)
//   bBlkShift <  0 : element = B[k*ldb + c]                       (row-major)
// z-slicing (grid.z): A += z*aZoff, B += z*bZoff, C += z*cZoff (hebbian heads)
// Requires: M % 128 == 0, Nc % 128 == 0, Kd % 32 == 0.
// ---------------------------------------------------------------------------
__global__ void gemm_wmma_f32(const float* __restrict__ A, int lda, int aZoff,
                              const float* __restrict__ B, int ldb, long bZoff,
                              int bBlkShift, long bBlkStride,
                              float* __restrict__ C, int ldc, int cZoff,
                              int M, int Nc, int Kd, int relu)
{
    __shared__ float Bs[128 * 34];       // B chunk, transposed + padded (17 KB)

    const int tid  = threadIdx.x;
    const int lane = tid & 31;
    const int wave = tid >> 5;
    const int half = lane >> 4;          // 0: K=0,1 / M+0 ; 1: K=2,3 / M+8
    const int l15  = lane & 15;

    const int z = blockIdx.z;
    A += (long)z * aZoff;
    B += (long)z * bZoff;
    C += (long)z * cZoff;

    const int blocksM = M >> 7;          // 128-row tiles
    const int bm = blockIdx.x % blocksM;
    const int bn = blockIdx.x / blocksM;
    const int row0 = (bm << 7) + (wave << 4);
    const int col0 = bn << 7;

    v8f acc[8];
    #pragma unroll
    for (int i = 0; i < 8; ++i) acc[i] = (v8f){0.f,0.f,0.f,0.f,0.f,0.f,0.f,0.f};

    // Per-lane A row pointer (A fragment: row = row0+l15, k = ka, ka+1)
    const float* aRow = A + (long)(row0 + l15) * lda;

    // Staging map: thread -> (column cS in strip, k rows kS0..kS0+15)
    const int cS  = tid & 127;
    const int kS0 = (tid >> 7) << 4;     // 0, 16
    const float* bStage;
    {
        int c = col0 + cS;
        if (bBlkShift >= 0) {
            int blk = c >> bBlkShift;
            int cc  = c & ((1 << bBlkShift) - 1);
            bStage = B + (long)blk * bBlkStride + cc;
        } else {
            bStage = B + c;
        }
    }

    for (int k0 = 0; k0 < Kd; k0 += 32) {
        // ---- stage B chunk (coalesced across 128 consecutive columns) ----
        #pragma unroll
        for (int r = 0; r < 16; ++r) {
            const int kk = kS0 + r;
            Bs[cS * 34 + kk] = bStage[(long)(k0 + kk) * ldb];
        }
        if (k0 + 32 < Kd)                 // prefetch next chunk into caches
            __builtin_prefetch(&bStage[(long)(k0 + 32 + kS0) * ldb], 0, 0);
        __syncthreads();

        // ---- 8 WMMA K-steps from LDS; A fragment amortized over 8 WMMAs ----
        #pragma unroll
        for (int s = 0; s < 8; ++s) {
            const int ka = (s << 2) + (half << 1);
            v2f a = *(const v2f*)(aRow + k0 + ka);         // 8B-aligned
            #pragma unroll
            for (int i = 0; i < 8; ++i) {
                v2f bf = *(const v2f*)(&Bs[(i * 16 + l15) * 34 + ka]);
                acc[i] = __builtin_amdgcn_wmma_f32_16x16x4_f32(
                    false, a, false, bf, (short)0, acc[i], false, false);
            }
        }
        __syncthreads();
    }

    // C/D layout: VGPR r holds C[row0 + r + 8*half][col0 + i*16 + l15]
    #pragma unroll
    for (int i = 0; i < 8; ++i) {
        const int col = col0 + i * 16 + l15;
        #pragma unroll
        for (int r = 0; r < 8; ++r) {
            const int row = row0 + r + (half << 3);
            float val = acc[i][r];
            if (relu) val = fmaxf(val, 0.f);
            C[(long)row * ldc + col] = val;
        }
    }
}

// ---------------------------------------------------------------------------
// sig[b,c] = mean_t q[b,t,c] + 0.5 * mean_{t%8==0} q[b,t,c]    (2 x 1024)
// ---------------------------------------------------------------------------
__global__ void sig_kernel(const float* __restrict__ q, float* __restrict__ sig)
{
    int idx = blockIdx.x * blockDim.x + threadIdx.x;   // 2048 threads
    if (idx >= 2 * 1024) return;
    int b = idx >> 10, c = idx & 1023;
    const float* p = q + (long)b * 1024 * 1024 + c;
    float s = 0.f, s8 = 0.f;
    for (int t = 0; t < 1024; ++t) {
        float v = p[(long)t * 1024];
        s += v;
        if ((t & 7) == 0) s8 += v;
    }
    sig[idx] = s * (1.f / 1024.f) + 0.5f * (s8 * (1.f / 128.f));
}

// h[b,j] = gelu_exact(sig[b,:] @ w1[:,j] + b1[j])      (2 x 512)
__global__ void mlp1_kernel(const float* __restrict__ sig,
                            const float* __restrict__ w1,
                            const float* __restrict__ b1,
                            float* __restrict__ h)
{
    int idx = blockIdx.x * blockDim.x + threadIdx.x;   // 1024 threads
    if (idx >= 2 * 512) return;
    int b = idx >> 9, j = idx & 511;
    const float* sp = sig + (long)b * 1024;
    float s = b1[j];
    for (int c = 0; c < 1024; ++c) s += sp[c] * w1[(long)c * 512 + j];
    h[idx] = 0.5f * s * (1.f + erff(s * 0.70710678118654752440f));
}

// mdiag[b,i] = softplus(base[i] + 0.1*(h[b,:] @ w2[:,i] + b2[i])) + 1e-6   (2 x 1024)
__global__ void mdiag_kernel(const float* __restrict__ h,
                             const float* __restrict__ w2,
                             const float* __restrict__ b2,
                             const float* __restrict__ base,
                             float* __restrict__ md)
{
    int idx = blockIdx.x * blockDim.x + threadIdx.x;   // 2048 threads
    if (idx >= 2 * 1024) return;
    int b = idx >> 10, i = idx & 1023;
    const float* hp = h + (long)b * 512;
    float s = b2[i];
    for (int j = 0; j < 512; ++j) s += hp[j] * w2[(long)j * 1024 + i];
    float x = base[i] + 0.1f * s;
    float sp = (x > 20.f) ? x : log1pf(expf(x));
    md[idx] = sp + 1e-6f;
}

// ---------------------------------------------------------------------------
// Sliding-window geometric interaction. One 128-thread block per (b,t,n).
// Phase 1: thread (k = tid>>2, p = tid&3) computes a 32-dim partial of
//   dist_sq(k) = sum_d mdiag[d]*(q[t,d]-q[t-31+k,d])^2   (padded rows => qw=0)
// Phase 2: thread d accumulates ctx[d] = sum_k w_k * v[t-31+k, d] / sum_k w_k
// ---------------------------------------------------------------------------
__global__ void context_kernel(const float* __restrict__ q,
                               const float* __restrict__ v,
                               const float* __restrict__ md,
                               float* __restrict__ ctx)
{
    const int g = blockIdx.x;            // ((b*T + t)*N + n)
    const int n = g & 7;
    const int bt = g >> 3;
    const int t = bt & 1023;
    const int b = bt >> 10;
    const int tid = threadIdx.x;

    __shared__ float qs[128];
    __shared__ float ms[128];
    __shared__ float part[32][4];
    __shared__ float wk[32];

    const long base = (long)g * 128;
    qs[tid] = q[base + tid];
    ms[tid] = md[(long)(b * 8 + n) * 128 + tid];
    __syncthreads();

    const int k = tid >> 2, p = tid & 3;
    const int ts = t + k - 31;
    float s = 0.f;
    if (ts >= 0) {
        const long wb = ((long)(b * 1024 + ts) * 8 + n) * 128;
        #pragma unroll 8
        for (int d = p * 32; d < p * 32 + 32; ++d) {
            float df = qs[d] - q[wb + d];
            s += ms[d] * df * df;
        }
    } else {
        #pragma unroll 8
        for (int d = p * 32; d < p * 32 + 32; ++d) {
            float df = qs[d];
            s += ms[d] * df * df;
        }
    }
    part[k][p] = s;
    __syncthreads();

    if (tid < 32) {
        float ds = part[tid][0] + part[tid][1] + part[tid][2] + part[tid][3];
        wk[tid] = expf(-sqrtf(ds + 1e-8f));
    }
    __syncthreads();

    float wsum = 0.f;
    #pragma unroll
    for (int i = 0; i < 32; ++i) wsum += wk[i];
    const float inv = 1.f / (wsum + 1e-8f);

    float acc = 0.f;
    for (int kk = 0; kk < 32; ++kk) {
        const int ts2 = t + kk - 31;
        if (ts2 >= 0) {
            const long wb = ((long)(b * 1024 + ts2) * 8 + n) * 128;
            acc += wk[kk] * v[wb + tid];
        }
    }
    ctx[base + tid] = acc * inv;
}

// qn = q / max(||q||_2 over hd, 1e-12) ; one block per (b,t,n), 128 threads
__global__ void qnorm_kernel(const float* __restrict__ q, float* __restrict__ qn)
{
    const int g = blockIdx.x;
    const int tid = threadIdx.x;
    const long base = (long)g * 128;
    __shared__ float red[128];
    float x = q[base + tid];
    red[tid] = x * x;
    __syncthreads();
    for (int s = 64; s > 0; s >>= 1) {
        if (tid < s) red[tid] += red[tid + s];
        __syncthreads();
    }
    float inv = 1.f / fmaxf(sqrtf(red[0]), 1e-12f);
    qn[base + tid] = x * inv;
}

// g = q * (ctx + heb)   elementwise
__global__ void combine_kernel(const float* __restrict__ q,
                               const float* __restrict__ ctx,
                               const float* __restrict__ heb,
                               float* __restrict__ g, long nelem)
{
    long i = (long)blockIdx.x * blockDim.x + threadIdx.x;
    if (i < nelem) g[i] = q[i] * (ctx[i] + heb[i]);
}

// ---------------------------------------------------------------------------
extern "C" void kernel_launch(void* const* d_in, const int* in_sizes, int n_in,
                              void* d_out, int out_size, void* d_ws, size_t ws_size,
                              hipStream_t stream)
{
    (void)in_sizes; (void)n_in; (void)out_size; (void)ws_size;

    const float* x        = (const float*)d_in[0];   // (2,1024,1024)
    const float* enc_q    = (const float*)d_in[1];   // (8,1024,128)
    const float* enc_v    = (const float*)d_in[2];   // (8,1024,128)
    const float* decoder  = (const float*)d_in[3];   // (1024,1024)
    const float* hebbian  = (const float*)d_in[4];   // (8,128,128)
    const float* m_w1     = (const float*)d_in[5];   // (1024,512)
    const float* m_b1     = (const float*)d_in[6];   // (512,)
    const float* m_w2     = (const float*)d_in[7];   // (512,1024)
    const float* m_b2     = (const float*)d_in[8];   // (1024,)
    const float* base_m   = (const float*)d_in[9];   // (8,128)
    float* out = (float*)d_out;                      // (2,1024,1024)

    const int M = 2048;          // B*T
    const int D = 1024;
    const int NHD = 1024;        // N*HD
    const long CH = (long)M * NHD;   // 2,097,152 elements per 8MB buffer

    float* ws  = (float*)d_ws;
    float* Q   = ws;             // q (relu)     2048x1024
    float* V   = ws + CH;        // v (relu)
    float* CTX = ws + 2 * CH;    // context
    float* QN  = ws + 3 * CH;    // normalized q
    float* HEB = ws + 4 * CH;    // hebbian context
    float* G   = ws + 5 * CH;    // q*(ctx+heb)
    float* SIG = ws + 6 * CH;    // 2048
    float* H   = SIG + 2048;     // 1024
    float* MD  = H + 1024;       // 2048

    dim3 blk(256);

    // q = relu(x @ enc_q)  — B in (N,D,HD) blocked-column layout
    const int blocksBig = (M / 128) * (NHD / 128);          // 16*8 = 128
    gemm_wmma_f32<<<dim3(blocksBig, 1, 1), blk, 0, stream>>>(
        x, D, 0, enc_q, 128, 0, /*shift*/7, (long)D * 128,
        Q, NHD, 0, M, NHD, D, /*relu*/1);
    // v = relu(x @ enc_v)
    gemm_wmma_f32<<<dim3(blocksBig, 1, 1), blk, 0, stream>>>(
        x, D, 0, enc_v, 128, 0, 7, (long)D * 128,
        V, NHD, 0, M, NHD, D, 1);

    // signature + modulation MLP + metric diag
    sig_kernel<<<8, 256, 0, stream>>>(Q, SIG);
    mlp1_kernel<<<4, 256, 0, stream>>>(SIG, m_w1, m_b1, H);
    mdiag_kernel<<<8, 256, 0, stream>>>(H, m_w2, m_b2, base_m, MD);

    // sliding-window context
    context_kernel<<<16384, 128, 0, stream>>>(Q, V, MD, CTX);

    // hebbian branch: qn, then per-head GEMM (grid.z = 8 heads)
    qnorm_kernel<<<16384, 128, 0, stream>>>(Q, QN);
    const int blocksHeb = (M / 128) * (128 / 128);          // 16
    gemm_wmma_f32<<<dim3(blocksHeb, 1, 8), blk, 0, stream>>>(
        QN, NHD, /*aZoff cols*/128, hebbian, 128, /*bZoff*/(long)128 * 128,
        /*shift*/-1, 0, HEB, NHD, /*cZoff cols*/128, M, 128, 128, 0);

    // g = q * (ctx + heb)
    combine_kernel<<<(int)((CH + 255) / 256), 256, 0, stream>>>(Q, CTX, HEB, G, CH);

    // out = g @ decoder
    gemm_wmma_f32<<<dim3(blocksBig, 1, 1), blk, 0, stream>>>(
        G, NHD, 0, decoder, D, 0, -1, 0,
        out, D, 0, M, D, NHD, 0);
}